// MonotonicOracle_66168266162602
// MI455X (gfx1250) — compile-verified
//
#include <hip/hip_runtime.h>
#include <hip/hip_bf16.h>
#include <math.h>

#define DDIM 64

typedef __attribute__((ext_vector_type(2))) float v2f;
typedef __attribute__((ext_vector_type(8))) float v8f;

__device__ __forceinline__ void atomAddF(float* p, float v) {
    unsafeAtomicAdd(p, v);
}

// ---------------- init: deg = 1 (self loop), accumulators = 0 ----------------
__global__ void mo_init_kernel(float* deg, float* wsum, float* scal, int N) {
    int i = blockIdx.x * blockDim.x + threadIdx.x;
    if (i < N)  deg[i]  = 1.0f;
    if (i < 64) wsum[i] = 0.0f;
    if (i < 8)  scal[i] = 0.0f;
}

// ---------------- h = node_emb + x @ score_w + score_b ----------------
__global__ void mo_h_kernel(const float* __restrict__ x, const float* __restrict__ emb,
                            const float* __restrict__ sw, const float* __restrict__ sb,
                            float* __restrict__ h, int N) {
    int i = blockIdx.x * blockDim.x + threadIdx.x;
    if (i >= N * DDIM) return;
    int n = i >> 6, d = i & 63;
    h[i] = emb[i] + x[2 * n] * sw[d] + x[2 * n + 1] * sw[DDIM + d] + sb[d];
}

// ---------------- hw = h @ gcn_w via V_WMMA_F32_16X16X4_F32 ----------------
// One wave32 computes one 16x16 output tile, accumulating over K=64 in steps of 4.
// A (16x4 f32): lane m=lane&15 holds row m; VGPR0 = K {0|2}, VGPR1 = K {1|3} by lane half.
// B (4x16 f32): lane holds column n=lane&15; VGPR0 = K {0|2}, VGPR1 = K {1|3} by lane half.
// C/D (16x16 f32): VGPR v, lanes 0-15 -> M=v, lanes 16-31 -> M=v+8, N=lane&15.
__global__ void mo_gemm_wmma(const float* __restrict__ h, const float* __restrict__ W,
                             float* __restrict__ hw, int nRowTiles) {
    int wave = (blockIdx.x * blockDim.x + threadIdx.x) >> 5;
    int lane = threadIdx.x & 31;
    int rowTile = wave >> 2;
    int colTile = wave & 3;
    if (rowTile >= nRowTiles) return;   // wave-uniform: active waves keep EXEC all-ones
    int row0 = rowTile << 4;
    int col0 = colTile << 4;
    int m = lane & 15;
    int khalf = (lane >> 4) << 1;       // 0 for lanes 0-15, 2 for lanes 16-31
    const float* arow = h + (size_t)(row0 + m) * DDIM;
    v8f acc = {};
#pragma unroll
    for (int k0 = 0; k0 < DDIM; k0 += 4) {
        v2f a = *(const v2f*)(arow + k0 + khalf);
        v2f b;
        b.x = W[(k0 + khalf) * DDIM + col0 + m];
        b.y = W[(k0 + khalf + 1) * DDIM + col0 + m];
        acc = __builtin_amdgcn_wmma_f32_16x16x4_f32(
            /*neg_a=*/false, a, /*neg_b=*/false, b,
            /*c_mod=*/(short)0, acc, /*reuse_a=*/false, /*reuse_b=*/false);
    }
    int rbase = row0 + ((lane >> 4) << 3);   // +8 for upper lane half
#pragma unroll
    for (int v = 0; v < 8; ++v)
        hw[(size_t)(rbase + v) * DDIM + col0 + m] = acc[v];
}

// ---------------- degree accumulation over dst ----------------
__global__ void mo_deg_kernel(const int* __restrict__ ei, float* __restrict__ deg, int E) {
    int e = blockIdx.x * blockDim.x + threadIdx.x;
    if (e >= E) return;
    atomAddF(&deg[ei[E + e]], 1.0f);
}

__global__ void mo_dinv_kernel(const float* __restrict__ deg, float* __restrict__ dinv, int N) {
    int n = blockIdx.x * blockDim.x + threadIdx.x;
    if (n < N) dinv[n] = rsqrtf(deg[n]);   // deg >= 1 always (self loop)
}

__global__ void mo_norm_kernel(const int* __restrict__ ei, const float* __restrict__ dinv,
                               float* __restrict__ nrm, int E) {
    int e = blockIdx.x * blockDim.x + threadIdx.x;
    if (e >= E) return;
    nrm[e] = dinv[ei[e]] * dinv[ei[E + e]];
}

// ---------------- z_acc[dst][d] += hw[src][d] * norm[e] ----------------
__global__ void mo_scatter_kernel(const int* __restrict__ ei, const float* __restrict__ hw,
                                  const float* __restrict__ nrm, float* __restrict__ zacc, int E) {
    int gid = blockIdx.x * blockDim.x + threadIdx.x;
    int e = gid >> 6;
    if (e >= E) return;
    int d = gid & 63;
    int s = ei[e], t = ei[E + e];
    atomAddF(&zacc[(size_t)t * DDIM + d], hw[(size_t)s * DDIM + d] * nrm[e]);
}

// ---------------- z = relu(zacc + hw*dinv^2 + b); fused reductions ----------------
// block = 256 threads = 64 dims x 4 node-subsets, 64 nodes per block.
__global__ void mo_zreduce_kernel(const float* __restrict__ zacc, const float* __restrict__ hw,
                                  const float* __restrict__ dinv, const float* __restrict__ gb,
                                  const float* __restrict__ x, const float* __restrict__ mask,
                                  const float* __restrict__ dw, float* __restrict__ wsum,
                                  float* __restrict__ scal, int N, const int* __restrict__ tgt) {
    __shared__ float sd[256];
    int tid = threadIdx.x;
    int d   = tid & 63;
    int sub = tid >> 6;
    int base = blockIdx.x * 64;
    int lim  = min(base + 64, N);
    int target = tgt[0];
    float bd = gb[d];
    float wacc = 0.0f, spacc = 0.0f;
    for (int n = base + sub; n < lim; n += 4) {
        float di = dinv[n];
        float z = zacc[(size_t)n * DDIM + d] + hw[(size_t)n * DDIM + d] * di * di + bd;
        z = fmaxf(z, 0.0f);
        wacc  += z * x[2 * n + 1];                       // scores = x[:,1]
        spacc += (z + log1pf(expf(-z))) * mask[n];       // softplus(z>=0)
        if (n == target) atomAddF(&scal[0], z * dw[d]);  // target_emb @ diff_w
    }
    sd[tid] = wacc;
    __syncthreads();
    if (sub == 0) {
        float w = sd[d] + sd[d + 64] + sd[d + 128] + sd[d + 192];
        atomAddF(&wsum[d], w);
    }
    __syncthreads();
    sd[tid] = spacc;
    __syncthreads();
    for (int s = 128; s > 0; s >>= 1) {
        if (tid < s) sd[tid] += sd[tid + s];
        __syncthreads();
    }
    if (tid == 0) atomAddF(&scal[1], sd[0]);
}

// ---------------- n_learned = sum x[:,0] ----------------
__global__ void mo_nlearned_kernel(const float* __restrict__ x, float* __restrict__ scal, int N) {
    __shared__ float sd[256];
    int tid = threadIdx.x;
    float acc = 0.0f;
    for (int n = blockIdx.x * 256 + tid; n < N; n += gridDim.x * 256)
        acc += x[2 * n];
    sd[tid] = acc;
    __syncthreads();
    for (int s = 128; s > 0; s >>= 1) {
        if (tid < s) sd[tid] += sd[tid + s];
        __syncthreads();
    }
    if (tid == 0) atomAddF(&scal[2], sd[0]);
}

// ---------------- finalize: ability, difficulty, prereq, gap, prob ----------------
__global__ void mo_final_kernel(const float* __restrict__ wsum, const float* __restrict__ scal,
                                const float* __restrict__ aw, const float* __restrict__ ab,
                                const float* __restrict__ db, const float* __restrict__ pw,
                                float* __restrict__ out) {
    if (threadIdx.x != 0 || blockIdx.x != 0) return;
    float nl = fmaxf(scal[2], 1.0f);
    float ability = ab[0];
    for (int d = 0; d < DDIM; ++d)
        ability += (wsum[d] / nl) * aw[d];
    float difficulty = scal[0] + db[0];
    float prereq = fabsf(pw[0]) * (scal[1] / (float)DDIM);
    float gap = ability - difficulty + prereq;
    out[0] = 1.0f / (1.0f + expf(-gap));
    out[1] = gap;
}

extern "C" void kernel_launch(void* const* d_in, const int* in_sizes, int n_in,
                              void* d_out, int out_size, void* d_ws, size_t ws_size,
                              hipStream_t stream) {
    const float* x    = (const float*)d_in[0];
    const float* mask = (const float*)d_in[1];
    const float* emb  = (const float*)d_in[2];
    const float* sw   = (const float*)d_in[3];
    const float* sb   = (const float*)d_in[4];
    const float* gw   = (const float*)d_in[5];
    const float* gb   = (const float*)d_in[6];
    const float* dw   = (const float*)d_in[7];
    const float* db   = (const float*)d_in[8];
    const float* aw   = (const float*)d_in[9];
    const float* ab   = (const float*)d_in[10];
    const float* pw   = (const float*)d_in[11];
    const int*   ei   = (const int*)d_in[12];
    const int*   tgt  = (const int*)d_in[13];
    float* out = (float*)d_out;

    int N = in_sizes[1];        // current_prereq_mask has N elements
    int E = in_sizes[12] / 2;   // edge_index is (2, E)

    // workspace carve-up (floats)
    float* ws = (float*)d_ws;
    size_t off = 0;
    float* h    = ws + off; off += (size_t)N * DDIM;   // reused as zacc after GEMM
    float* hw   = ws + off; off += (size_t)N * DDIM;
    float* deg  = ws + off; off += (size_t)N;
    float* dinv = ws + off; off += (size_t)N;
    float* nrm  = ws + off; off += (size_t)E;
    float* wsum = ws + off; off += 64;
    float* scal = ws + off; off += 8;   // [0]=diff_acc [1]=sp_sum [2]=n_learned

    const int T = 256;
    int nRowTiles = (N + 15) / 16;
    int waves     = nRowTiles * 4;

    mo_init_kernel<<<(N + T - 1) / T, T, 0, stream>>>(deg, wsum, scal, N);
    mo_h_kernel<<<((size_t)N * DDIM + T - 1) / T, T, 0, stream>>>(x, emb, sw, sb, h, N);
    mo_gemm_wmma<<<((size_t)waves * 32 + T - 1) / T, T, 0, stream>>>(h, gw, hw, nRowTiles);
    mo_deg_kernel<<<(E + T - 1) / T, T, 0, stream>>>(ei, deg, E);
    mo_dinv_kernel<<<(N + T - 1) / T, T, 0, stream>>>(deg, dinv, N);
    mo_norm_kernel<<<(E + T - 1) / T, T, 0, stream>>>(ei, dinv, nrm, E);
    // h is fully consumed by the GEMM above (stream order); reuse it as zacc
    hipMemsetAsync(h, 0, (size_t)N * DDIM * sizeof(float), stream);
    mo_scatter_kernel<<<((size_t)E * DDIM + T - 1) / T, T, 0, stream>>>(ei, hw, nrm, h, E);
    mo_zreduce_kernel<<<(N + 63) / 64, T, 0, stream>>>(h, hw, dinv, gb, x, mask, dw,
                                                       wsum, scal, N, tgt);
    mo_nlearned_kernel<<<64, T, 0, stream>>>(x, scal, N);
    mo_final_kernel<<<1, 64, 0, stream>>>(wsum, scal, aw, ab, db, pw, out);
}